// GMAttendSoftmax4_28028956573868
// MI455X (gfx1250) — compile-verified
//
#include <hip/hip_runtime.h>
#include <math.h>

// ---------------- problem constants ----------------
#define Bb 4
#define Nn 2048          // N == M == 2048
#define Hh 512
#define Kk 64            // KEY_DIM
#define NP 2049          // padded (slack) dimension
#define ST 2064          // row stride of padded log-matrix (16-float aligned)
#define NITERS 20
#define EPS_IN 1e-5f

typedef __attribute__((ext_vector_type(2))) float v2f;
typedef __attribute__((ext_vector_type(8))) float v8f;

// ---------------- small utility kernels ----------------
__global__ void zero_stats_kernel(float* stats) {
    if (threadIdx.x < 16) stats[threadIdx.x] = 0.0f;
}

// out[r][k] = dot(X[r, :], W[k, :]) + bq[k]   (rows = B*N, k in [0,64))
__global__ __launch_bounds__(256) void proj_kernel(const float* __restrict__ X,
                                                   const float* __restrict__ W,
                                                   const float* __restrict__ bq,
                                                   float* __restrict__ out) {
    int idx = blockIdx.x * 256 + threadIdx.x;          // exactly B*N*64 threads
    int r = idx >> 6;
    int k = idx & 63;
    const float4* xr = (const float4*)(X + (size_t)r * Hh);
    const float4* wr = (const float4*)(W + (size_t)k * Hh);
    float acc = 0.0f;
#pragma unroll 4
    for (int h = 0; h < Hh / 4; ++h) {
        float4 a = xr[h];
        float4 c = wr[h];
        acc = fmaf(a.x, c.x, acc);
        acc = fmaf(a.y, c.y, acc);
        acc = fmaf(a.z, c.z, acc);
        acc = fmaf(a.w, c.w, acc);
    }
    out[idx] = acc + bq[k];
}

// norms[r] = sum_k M[r][k]^2
__global__ __launch_bounds__(256) void norm_kernel(const float* __restrict__ M,
                                                   float* __restrict__ norms, int rows) {
    int r = blockIdx.x * 256 + threadIdx.x;
    if (r >= rows) return;
    const float4* p = (const float4*)(M + (size_t)r * Kk);
    float s = 0.0f;
#pragma unroll
    for (int i = 0; i < Kk / 4; ++i) {
        float4 v = p[i];
        s = fmaf(v.x, v.x, s);
        s = fmaf(v.y, v.y, s);
        s = fmaf(v.z, v.z, s);
        s = fmaf(v.w, v.w, s);
    }
    norms[r] = s;
}

// ---------------- WMMA distance/exp kernel ----------------
// 128x128 output tile per block; 8 waves, wave w owns rows [16w,16w+16), all 128 cols.
// e = exp(2*acc - ||xw_n||^2 - ||yw_m||^2) written into interior of LA; stats += (sum, sumsq).
#define LDS_STRIDE 68
__global__ __launch_bounds__(256) void gemm_e_kernel(const float* __restrict__ XW,
                                                     const float* __restrict__ YW,
                                                     const float* __restrict__ XN,
                                                     const float* __restrict__ YN,
                                                     float* __restrict__ LA,
                                                     float* __restrict__ stats) {
    __shared__ float lx[128 * LDS_STRIDE];
    __shared__ float ly[128 * LDS_STRIDE];
    const int b = blockIdx.z;
    const int n0 = blockIdx.x * 128;
    const int m0 = blockIdx.y * 128;
    const int tid = threadIdx.x;

    const float4* xg = (const float4*)(XW + ((size_t)b * Nn + n0) * Kk);
    const float4* yg = (const float4*)(YW + ((size_t)b * Nn + m0) * Kk);
#pragma unroll
    for (int i = 0; i < 8; ++i) {              // 2048 float4 per tile
        int f = tid + i * 256;
        int row = f >> 4;                       // 16 float4 per 64-float row
        int kq = f & 15;
        float4 xv = xg[row * 16 + kq];
        float4 yv = yg[row * 16 + kq];
        float* dx = &lx[row * LDS_STRIDE + kq * 4];
        float* dy = &ly[row * LDS_STRIDE + kq * 4];
        dx[0] = xv.x; dx[1] = xv.y; dx[2] = xv.z; dx[3] = xv.w;
        dy[0] = yv.x; dy[1] = yv.y; dy[2] = yv.z; dy[3] = yv.w;
    }
    __syncthreads();

    const int wave = tid >> 5;
    const int lane = tid & 31;
    const int half = lane >> 4;                 // 0: K={k,k+1}, 1: K={k+2,k+3}
    const int l15 = lane & 15;
    const int arow = wave * 16 + l15;           // A fragment row (M = lane%16)
    const int kh = half * 2;

    v8f acc[8];
#pragma unroll
    for (int t = 0; t < 8; ++t) acc[t] = (v8f)(0.0f);

    for (int k = 0; k < Kk; k += 4) {
        v2f a;
        a.x = lx[arow * LDS_STRIDE + k + kh];
        a.y = lx[arow * LDS_STRIDE + k + kh + 1];
#pragma unroll
        for (int t = 0; t < 8; ++t) {
            int bcol = t * 16 + l15;            // B fragment col (N = lane%16)
            v2f bf;
            bf.x = ly[bcol * LDS_STRIDE + k + kh];
            bf.y = ly[bcol * LDS_STRIDE + k + kh + 1];
            acc[t] = __builtin_amdgcn_wmma_f32_16x16x4_f32(
                false, a, false, bf, (short)0, acc[t], false, false);
        }
    }

    // epilogue: C layout -> VGPR v, lanes 0-15: row v; lanes 16-31: row v+8; col = lane%16
    float s = 0.0f, ss = 0.0f;
    float* lap = LA + (size_t)b * NP * ST;
    const int rbase = n0 + wave * 16 + half * 8;
    float xn[8];
#pragma unroll
    for (int v = 0; v < 8; ++v) xn[v] = XN[b * Nn + rbase + v];
#pragma unroll
    for (int t = 0; t < 8; ++t) {
        int col = m0 + t * 16 + l15;
        float yn = YN[b * Nn + col];
#pragma unroll
        for (int v = 0; v < 8; ++v) {
            float e = __expf(2.0f * acc[t][v] - xn[v] - yn);
            lap[(size_t)(rbase + v) * ST + col] = e;
            s += e;
            ss = fmaf(e, e, ss);
        }
    }

    __syncthreads();
    lx[tid] = s;
    lx[256 + tid] = ss;
    __syncthreads();
    for (int o = 128; o > 0; o >>= 1) {
        if (tid < o) { lx[tid] += lx[tid + o]; lx[256 + tid] += lx[256 + tid + o]; }
        __syncthreads();
    }
    if (tid == 0) {
        atomicAdd(&stats[b * 2 + 0], lx[0]);
        atomicAdd(&stats[b * 2 + 1], lx[256]);
    }
}

// stats[8+2b] = scale = in_w/sqrt(var+eps); stats[8+2b+1] = shift = in_b - mean*scale
__global__ void finalize_stats_kernel(float* stats, const float* __restrict__ in_w,
                                      const float* __restrict__ in_b) {
    int b = threadIdx.x;
    if (b >= Bb) return;
    float cnt = (float)Nn * (float)Nn;
    float mean = stats[b * 2 + 0] / cnt;
    float var = stats[b * 2 + 1] / cnt - mean * mean;
    float scale = in_w[0] * rsqrtf(var + EPS_IN);
    stats[8 + b * 2 + 0] = scale;
    stats[8 + b * 2 + 1] = in_b[0] - mean * scale;
}

// la = e*scale + shift in interior; zero slack row/col
__global__ __launch_bounds__(256) void build_la_kernel(float* __restrict__ LA,
                                                       const float* __restrict__ stats) {
    int idx = blockIdx.x * 256 + threadIdx.x;
    if (idx >= NP * NP) return;
    int b = blockIdx.y;
    int i = idx / NP;
    int j = idx - i * NP;
    float* lap = LA + (size_t)b * NP * ST;
    size_t a = (size_t)i * ST + j;
    if (i == Nn || j == Nn) {
        lap[a] = 0.0f;
    } else {
        lap[a] = lap[a] * stats[8 + b * 2] + stats[8 + b * 2 + 1];
    }
}

// ---------------- Sinkhorn passes ----------------
__device__ inline void lse_merge(float& m, float& s, float m2, float s2) {
    if (m2 > m) { s = s * __expf(m - m2) + s2; m = m2; }
    else        { s = s + s2 * __expf(m2 - m); }
}

// lse over all NP cols of row r (rows < Nn only)
__global__ __launch_bounds__(256) void row_lse_kernel(const float* __restrict__ LA,
                                                      float* __restrict__ LSER) {
    int r = blockIdx.x;
    int b = blockIdx.y;
    int tid = threadIdx.x;
    const float* rp = LA + (size_t)b * NP * ST + (size_t)r * ST;
    float m = -1e30f, s = 0.0f;
    for (int j = tid; j < NP; j += 256) {
        float v = rp[j];
        if (v > m) { s = s * __expf(m - v) + 1.0f; m = v; }
        else       { s += __expf(v - m); }
    }
    __shared__ float sm[256], ssv[256];
    sm[tid] = m; ssv[tid] = s;
    __syncthreads();
    for (int o = 128; o > 0; o >>= 1) {
        if (tid < o) {
            float mm = sm[tid], scur = ssv[tid];
            lse_merge(mm, scur, sm[tid + o], ssv[tid + o]);
            sm[tid] = mm; ssv[tid] = scur;
        }
        __syncthreads();
    }
    if (tid == 0) LSER[b * Nn + r] = sm[0] + __logf(ssv[0]);
}

// lse over all NP rows of col c (cols < Nn), on row-normalized values.
// block = 32 cols x 8 row-chunks
__global__ __launch_bounds__(256) void col_lse_kernel(const float* __restrict__ LA,
                                                      const float* __restrict__ LSER,
                                                      float* __restrict__ LSEC) {
    int tid = threadIdx.x;
    int b = blockIdx.y;
    int c = blockIdx.x * 32 + (tid & 31);
    int chunk = tid >> 5;
    const float* base = LA + (size_t)b * NP * ST;
    float m = -1e30f, s = 0.0f;
    int i0 = chunk * 257;
    int i1 = i0 + 257; if (i1 > NP) i1 = NP;
    for (int i = i0; i < i1; ++i) {
        float v = base[(size_t)i * ST + c];
        if (i < Nn) v -= LSER[b * Nn + i];
        if (v > m) { s = s * __expf(m - v) + 1.0f; m = v; }
        else       { s += __expf(v - m); }
    }
    __shared__ float sm[256], ssv[256];
    sm[tid] = m; ssv[tid] = s;
    __syncthreads();
    if (chunk == 0) {
        float mm = sm[tid], scur = ssv[tid];
#pragma unroll
        for (int q = 1; q < 8; ++q) lse_merge(mm, scur, sm[tid + 32 * q], ssv[tid + 32 * q]);
        LSEC[b * Nn + c] = mm + __logf(scur);
    }
}

// la[i][j] -= (i<Nn ? lse_row[i] : 0) + (j<Nn ? lse_col[j] : 0)
__global__ __launch_bounds__(256) void update_la_kernel(float* __restrict__ LA,
                                                        const float* __restrict__ LSER,
                                                        const float* __restrict__ LSEC) {
    int idx = blockIdx.x * 256 + threadIdx.x;
    if (idx >= NP * NP) return;
    int b = blockIdx.y;
    int i = idx / NP;
    int j = idx - i * NP;
    float* lap = LA + (size_t)b * NP * ST;
    size_t a = (size_t)i * ST + j;
    float v = lap[a];
    if (i < Nn) v -= LSER[b * Nn + i];
    if (j < Nn) v -= LSEC[b * Nn + j];
    lap[a] = v;
}

// w[b,r] = exp(max_{j<Nn} la[b,r,j])
__global__ __launch_bounds__(256) void rowmax_kernel(const float* __restrict__ LA,
                                                     float* __restrict__ WV) {
    int r = blockIdx.x;
    int b = blockIdx.y;
    int tid = threadIdx.x;
    const float* rp = LA + (size_t)b * NP * ST + (size_t)r * ST;
    float m = -1e30f;
    for (int j = tid; j < Nn; j += 256) { float v = rp[j]; m = v > m ? v : m; }
    __shared__ float sm[256];
    sm[tid] = m;
    __syncthreads();
    for (int o = 128; o > 0; o >>= 1) {
        if (tid < o) sm[tid] = sm[tid] > sm[tid + o] ? sm[tid] : sm[tid + o];
        __syncthreads();
    }
    if (tid == 0) WV[b * Nn + r] = __expf(sm[0]);
}

// out[b,i,j] = w[b,i] * w[b,j]
__global__ __launch_bounds__(256) void outer_kernel(const float* __restrict__ WV,
                                                    float* __restrict__ out) {
    int idx = blockIdx.x * 256 + threadIdx.x;
    int b = blockIdx.y;
    int i = idx >> 11;          // /2048
    int j = idx & 2047;
    out[(size_t)b * Nn * Nn + idx] = WV[b * Nn + i] * WV[b * Nn + j];
}

// ---------------- launch ----------------
extern "C" void kernel_launch(void* const* d_in, const int* in_sizes, int n_in,
                              void* d_out, int out_size, void* d_ws, size_t ws_size,
                              hipStream_t stream) {
    const float* x    = (const float*)d_in[0];
    const float* y    = (const float*)d_in[1];
    const float* Wq   = (const float*)d_in[2];
    const float* bq   = (const float*)d_in[3];
    const float* in_w = (const float*)d_in[4];
    const float* in_b = (const float*)d_in[5];
    float* out = (float*)d_out;

    float* ws = (float*)d_ws;
    float* XW    = ws;                    // B*N*K
    float* YW    = XW + (size_t)Bb * Nn * Kk;
    float* XN    = YW + (size_t)Bb * Nn * Kk;   // B*N
    float* YN    = XN + Bb * Nn;
    float* LSER  = YN + Bb * Nn;
    float* LSEC  = LSER + Bb * Nn;
    float* WV    = LSEC + Bb * Nn;
    float* STATS = WV + Bb * Nn;          // 16 floats
    float* LA    = STATS + 64;            // B * NP * ST

    zero_stats_kernel<<<1, 32, 0, stream>>>(STATS);

    // projections: exactly B*N*64 threads each
    proj_kernel<<<(Bb * Nn * Kk) / 256, 256, 0, stream>>>(x, Wq, bq, XW);
    proj_kernel<<<(Bb * Nn * Kk) / 256, 256, 0, stream>>>(y, Wq, bq, YW);
    norm_kernel<<<(Bb * Nn + 255) / 256, 256, 0, stream>>>(XW, XN, Bb * Nn);
    norm_kernel<<<(Bb * Nn + 255) / 256, 256, 0, stream>>>(YW, YN, Bb * Nn);

    // WMMA cross-term + exp + stats
    dim3 gg(Nn / 128, Nn / 128, Bb);
    gemm_e_kernel<<<gg, 256, 0, stream>>>(XW, YW, XN, YN, LA, STATS);

    finalize_stats_kernel<<<1, 32, 0, stream>>>(STATS, in_w, in_b);

    dim3 gfull((NP * NP + 255) / 256, Bb);
    build_la_kernel<<<gfull, 256, 0, stream>>>(LA, STATS);

    for (int it = 0; it < NITERS; ++it) {
        row_lse_kernel<<<dim3(Nn, Bb), 256, 0, stream>>>(LA, LSER);
        col_lse_kernel<<<dim3(Nn / 32, Bb), 256, 0, stream>>>(LA, LSER, LSEC);
        update_la_kernel<<<gfull, 256, 0, stream>>>(LA, LSER, LSEC);
    }

    rowmax_kernel<<<dim3(Nn, Bb), 256, 0, stream>>>(LA, WV);
    outer_kernel<<<dim3((Nn * Nn) / 256, Bb), 256, 0, stream>>>(WV, out);
}